// decompose_emb_85727547228275
// MI455X (gfx1250) — compile-verified
//
#include <hip/hip_runtime.h>
#include <stdint.h>

#define B_  32
#define L_  4096
#define C_  512
#define P_  128
#define NP_ 32
#define D_  512

typedef __attribute__((ext_vector_type(2))) float v2f;
typedef __attribute__((ext_vector_type(8))) float v8f;

// ---------------------------------------------------------------------------
// CDNA5 async global->LDS copy (bypasses VGPRs, tracked by ASYNCcnt).
// GVS mode: 64-bit SGPR base + 32-bit per-lane VGPR offset; dst is the
// 32-bit LDS offset (low 32 bits of the generic shared-memory pointer).
// ---------------------------------------------------------------------------
__device__ __forceinline__ void async_copy_b128(uint32_t lds_off,
                                                const void* gbase,
                                                uint32_t goff) {
    asm volatile("global_load_async_to_lds_b128 %0, %1, %2 offset:0"
                 :: "v"(lds_off), "v"(goff), "s"(gbase)
                 : "memory");
}
__device__ __forceinline__ void wait_async_le(void) {   // <= 16 outstanding
    asm volatile("s_wait_asynccnt 0x10" ::: "memory");
}
__device__ __forceinline__ void wait_async_0(void) {    // all done
    asm volatile("s_wait_asynccnt 0x0" ::: "memory");
}

// ---------------------------------------------------------------------------
// Kernel 1: EMA decomposition.
// grid = B * (C/32) blocks, 32 threads (one wave32). lane == channel offset,
// so x[b, l, c0+lane] loads are contiguous 128B. Trend/season are transposed
// through LDS so global writes are contiguous along L.
// season is stashed at out[m*512 + p] (first 128 cols of each output row),
// m = (b*C + c)*NP + n, p = l % 128  (note (b*C+c)*L + l == m*128 + p).
// ---------------------------------------------------------------------------
__global__ __launch_bounds__(32)
void ema_kernel(const float* __restrict__ x,
                const float* __restrict__ alpha,
                float* __restrict__ season_stash,   // rows of 512, cols 0..127 used
                float* __restrict__ trend)          // [B*C, L]
{
    __shared__ float lds[32][33];
    const int lane = threadIdx.x;          // channel within tile
    const int b    = blockIdx.x >> 4;      // / (C/32)
    const int ct   = blockIdx.x & 15;
    const int c    = ct * 32 + lane;

    const float a  = alpha[c];
    const float om = 1.0f - a;
    float t = 0.0f;

    const size_t xbase = (size_t)b * L_ * C_ + c;

    for (int l0 = 0; l0 < L_; l0 += 32) {
        float tr[32], se[32];
#pragma unroll
        for (int j = 0; j < 32; ++j) {
            float xv = x[xbase + (size_t)(l0 + j) * C_];
            t = (l0 + j == 0) ? xv : fmaf(om, t, a * xv);
            tr[j] = t;
            se[j] = xv - t;
        }

        // ---- trend transpose + coalesced nontemporal store ----
#pragma unroll
        for (int j = 0; j < 32; ++j) lds[j][lane] = tr[j];
        __syncthreads();
#pragma unroll
        for (int r = 0; r < 32; ++r) {
            size_t row = (size_t)(b * C_ + ct * 32 + r);
            __builtin_nontemporal_store(lds[lane][r], &trend[row * L_ + l0 + lane]);
        }
        __syncthreads();

        // ---- season transpose + coalesced stash (re-read by GEMM: keep RT) ----
#pragma unroll
        for (int j = 0; j < 32; ++j) lds[j][lane] = se[j];
        __syncthreads();
#pragma unroll
        for (int r = 0; r < 32; ++r) {
            size_t m = (size_t)(b * C_ + ct * 32 + r) * NP_ + (l0 >> 7);
            season_stash[m * D_ + (l0 & (P_ - 1)) + lane] = lds[lane][r];
        }
        __syncthreads();
    }
}

// ---------------------------------------------------------------------------
// Kernel 2: season_emb = A(524288x128) * W(128x512) with fp32 WMMA 16x16x4.
// A staged to LDS via async copy before any C write (in-place safe: each
// block owns its 128 rows). W panels are double-buffered: async prefetch of
// panel nb+1 overlaps compute on panel nb; s_wait_asynccnt 16 waits for the
// current panel only (async loads complete in order).
// ---------------------------------------------------------------------------
#define BM  128
#define LDA 132   // pad: stride 132 floats -> conflict-free b64 A-frag loads

__global__ __launch_bounds__(256)
void gemm_kernel(const float* __restrict__ W,   // [128, 512] row-major
                 float* __restrict__ out)       // [M, 512]; cols 0..127 hold A
{
    __shared__ float As[BM][LDA];         // 67584 B
    __shared__ float Bs[2][128][128];     // 2 x 65536 B  (total ~196 KB LDS)

    const int tid  = threadIdx.x;
    const int lane = tid & 31;
    const int wave = tid >> 5;
    const int wm   = wave >> 2;           // 0..1
    const int wn   = wave & 3;            // 0..3
    const size_t m0 = (size_t)blockIdx.x * BM;

    const int r0 = tid >> 5;              // 8 rows per staging pass
    const int c4 = (tid & 31) * 4;        // float4 column

    const uint32_t as_base  = (uint32_t)(size_t)&As[0][0];
    const uint32_t bs_base0 = (uint32_t)(size_t)&Bs[0][0][0];
    const uint32_t bs_base1 = (uint32_t)(size_t)&Bs[1][0][0];
    const float* abase = out + m0 * (size_t)D_;

    // ---- async stage A tile (128 x 128, stride 512) ----
#pragma unroll
    for (int p = 0; p < 16; ++p) {
        int r = p * 8 + r0;
        async_copy_b128(as_base + (uint32_t)(r * LDA + c4) * 4u,
                        abase, (uint32_t)(r * D_ + c4) * 4u);
    }
    // ---- async stage W panel 0 ----
#pragma unroll
    for (int p = 0; p < 16; ++p) {
        int r = p * 8 + r0;
        async_copy_b128(bs_base0 + (uint32_t)(r * 128 + c4) * 4u,
                        W, (uint32_t)(r * D_ + c4) * 4u);
    }

    const int am = lane & 15;             // M (or N) index within 16
    const int kh = (lane >> 4) * 2;       // K half select: 0 or 2

    int buf = 0;
    for (int nb = 0; nb < 4; ++nb) {
        // prefetch next W panel into the other buffer (its last readers
        // finished at the barrier ending the previous chunk)
        if (nb + 1 < 4) {
            uint32_t bb = (buf ^ 1) ? bs_base1 : bs_base0;
#pragma unroll
            for (int p = 0; p < 16; ++p) {
                int r = p * 8 + r0;
                async_copy_b128(bb + (uint32_t)(r * 128 + c4) * 4u,
                                W, (uint32_t)(r * D_ + (nb + 1) * 128 + c4) * 4u);
            }
            wait_async_le();              // current panel (and A) complete
        } else {
            wait_async_0();
        }
        __syncthreads();                  // all waves' async data visible

        const float (*Bp)[128] = Bs[buf];

        v8f z = {0.f, 0.f, 0.f, 0.f, 0.f, 0.f, 0.f, 0.f};
        v8f acc[4][2];
#pragma unroll
        for (int mt = 0; mt < 4; ++mt)
#pragma unroll
            for (int nt = 0; nt < 2; ++nt) acc[mt][nt] = z;

        for (int k = 0; k < 128; k += 4) {
            v2f af[4];
#pragma unroll
            for (int mt = 0; mt < 4; ++mt) {
                const float* ap = &As[wm * 64 + mt * 16 + am][k + kh];
                af[mt].x = ap[0];
                af[mt].y = ap[1];
            }
            v2f bf[2];
#pragma unroll
            for (int nt = 0; nt < 2; ++nt) {
                int col = wn * 32 + nt * 16 + am;
                bf[nt].x = Bp[k + kh][col];
                bf[nt].y = Bp[k + kh + 1][col];
            }
#pragma unroll
            for (int mt = 0; mt < 4; ++mt)
#pragma unroll
                for (int nt = 0; nt < 2; ++nt)
                    acc[mt][nt] = __builtin_amdgcn_wmma_f32_16x16x4_f32(
                        false, af[mt], false, bf[nt],
                        (short)0, acc[mt][nt], false, false);
        }

        // ---- nontemporal C store (write-once 1GB stream) ----
#pragma unroll
        for (int mt = 0; mt < 4; ++mt)
#pragma unroll
            for (int nt = 0; nt < 2; ++nt)
#pragma unroll
                for (int v = 0; v < 8; ++v) {
                    int row = wm * 64 + mt * 16 + v + ((lane >> 4) * 8);
                    int col = nb * 128 + wn * 32 + nt * 16 + (lane & 15);
                    __builtin_nontemporal_store(acc[mt][nt][v],
                        &out[(m0 + row) * (size_t)D_ + col]);
                }

        __syncthreads();                  // readers done before buf is reused
        buf ^= 1;
    }
}

// ---------------------------------------------------------------------------
extern "C" void kernel_launch(void* const* d_in, const int* in_sizes, int n_in,
                              void* d_out, int out_size, void* d_ws, size_t ws_size,
                              hipStream_t stream) {
    (void)in_sizes; (void)n_in; (void)out_size; (void)d_ws; (void)ws_size;

    const float* x     = (const float*)d_in[0];
    // d_in[1] = x_mark (unused by the reference computation)
    const float* alpha = (const float*)d_in[2];
    const float* W     = (const float*)d_in[3];

    float* out    = (float*)d_out;
    float* season = out;                                   // [M, 512]
    float* trend  = out + (size_t)B_ * C_ * NP_ * D_;      // [B*C, L]

    ema_kernel<<<dim3(B_ * (C_ / 32)), dim3(32), 0, stream>>>(x, alpha, season, trend);
    gemm_kernel<<<dim3((B_ * C_ * NP_) / BM), dim3(256), 0, stream>>>(W, season);
}